// HybridCoordinateBlockStepper_59322088292718
// MI455X (gfx1250) — compile-verified
//
#include <hip/hip_runtime.h>
#include <math.h>

typedef __attribute__((ext_vector_type(2))) float v2f;
typedef __attribute__((ext_vector_type(8))) float v8f;

#define R_GAS_C   287.0f
#define C_P_C     1004.0f
#define R_EARTH_C 6371000.0f
#define NU_C      100000.0f

#define KD   8
#define HD   180
#define WD   360
#define NTAP 7
#define KS_C 5
#define MD   181        // rfft bins
#define LDF  368        // padded spectral width (2*181=362 -> 368)
#define LDF2 184        // LDF in float2 units
#define NCOUT 98        // grad(50) + vec(16) + lap(32)
#define MROWS_F 7424    // 41*180=7380 padded to mult of 64
#define MROWS_I 17664   // 98*180=17640 padded to mult of 64

// ---- workspace layout (in floats) ----
static constexpr size_t OFF_FWD  = 0;                                // 360x368 (pair-interleaved)
static constexpr size_t OFF_INV  = OFF_FWD  + (size_t)360*368;       // 368x368 (pair-interleaved)
static constexpr size_t OFF_AIN  = OFF_INV  + (size_t)368*368;       // 7424x360
static constexpr size_t OFF_GHAT = OFF_AIN  + (size_t)MROWS_F*360;   // 7424x368
static constexpr size_t OFF_YS   = OFF_GHAT + (size_t)MROWS_F*LDF;   // 17664x368
static constexpr size_t OFF_OUTS = OFF_YS   + (size_t)MROWS_I*LDF;   // 17664x368
static constexpr size_t OFF_PV   = OFF_OUTS + (size_t)MROWS_I*LDF;   // 180*7*181*4
static constexpr size_t OFF_PVV  = OFF_PV   + (size_t)HD*NTAP*MD*4;  // *8
static constexpr size_t OFF_PS   = OFF_PVV  + (size_t)HD*NTAP*MD*8;  // *2
static constexpr size_t WS_FLOATS= OFF_PS   + (size_t)HD*NTAP*MD*2;

// ----------------------------------------------------------------------------
// DFT matrices, stored PAIR-INTERLEAVED along K so a WMMA B-fragment is one
// aligned b64 load: element (k,n) lives at [((k>>1)*LDN + n)*2 + (k&1)].
// FWD: K=w (360 samples), N=2m+p spectral (cos, -sin), unnormalized rfft.
// INV: K=2m+p spectral, N=w; (2-delta)/360 scaling; sin rows negated (irfft).
// Integer angle reduction (m*w) mod 360 keeps args in [0, 2pi) exactly.
// ----------------------------------------------------------------------------
__global__ void gen_dft(float* __restrict__ FWD, float* __restrict__ INV) {
  int t = blockIdx.x * blockDim.x + threadIdx.x;
  const float C = 0.017453292519943295f; // 2*pi/360
  if (t < 360 * 368) {
    int w = t / 368, col = t % 368;      // k = w, n = col
    float val = 0.f;
    if (col < 362) {
      int m = col >> 1, p = col & 1;
      int r = (m * w) % 360;
      float th = (float)r * C;
      val = p ? -__sinf(th) : __cosf(th);
    }
    FWD[((size_t)(w >> 1) * 368 + col) * 2 + (w & 1)] = val;
  }
  if (t < 368 * 368) {
    int j = t / 368, w = t % 368;        // k = j, n = w
    float val = 0.f;
    if (j < 362 && w < 360) {
      int m = j >> 1, p = j & 1;
      int r = (m * w) % 360;
      float th = (float)r * C;
      float s = ((m == 0) || (m == 180)) ? (1.0f / 360.0f) : (2.0f / 360.0f);
      val = (p ? -__sinf(th) : __cosf(th)) * s;
    }
    INV[((size_t)(j >> 1) * 368 + w) * 2 + (j & 1)] = val;
  }
}

// ----------------------------------------------------------------------------
// Collapse psi over k (exact: irfft and the k-contractions are linear).
// Folds 1/R into psi_v/psi_vv sums and NU/R^2 * lap_w into psi_s.
// ----------------------------------------------------------------------------
__global__ void psi_combine(const float2* __restrict__ psi_s,
                            const float2* __restrict__ psi_v,
                            const float2* __restrict__ psi_vv,
                            const float*  __restrict__ lap_w,
                            float2* __restrict__ pV, float2* __restrict__ pVV,
                            float2* __restrict__ pS) {
  int t = blockIdx.x * blockDim.x + threadIdx.x;
  const int NHM = HD * NTAP * MD;
  if (t >= NHM) return;
  const float invR = 1.0f / R_EARTH_C;
  const float lsc  = NU_C / (R_EARTH_C * R_EARTH_C);
#pragma unroll
  for (int d = 0; d < 2; ++d) {
    float re = 0.f, im = 0.f;
    for (int k = 0; k < KS_C; ++k) {
      float2 c = psi_v[(size_t)k * NHM * 2 + (size_t)t * 2 + d];
      re += c.x; im += c.y;
    }
    pV[(size_t)t * 2 + d] = make_float2(re * invR, im * invR);
  }
#pragma unroll
  for (int ed = 0; ed < 4; ++ed) {
    float re = 0.f, im = 0.f;
    for (int k = 0; k < KS_C; ++k) {
      float2 c = psi_vv[(size_t)k * NHM * 4 + (size_t)t * 4 + ed];
      re += c.x; im += c.y;
    }
    pVV[(size_t)t * 4 + ed] = make_float2(re * invR, im * invR);
  }
  {
    float re = 0.f, im = 0.f;
    for (int k = 0; k < KS_C; ++k) {
      float lw = lap_w[k];
      float2 c = psi_s[(size_t)k * NHM + t];
      re += lw * c.x; im += lw * c.y;
    }
    pS[t] = make_float2(re * lsc, im * lsc);
  }
}

// ----------------------------------------------------------------------------
// Pack the 41 forward channels; computes hydrostatic phi_mid + kinetic energy
// per column. Row r = c*180 + h, width 360.
// Channels: 0..7 phi+ke, 8..15 T, 16..23 q, 24 ps, 25..32 u, 33..40 v.
// ----------------------------------------------------------------------------
__global__ void pack_fields(const float* __restrict__ uv, const float* __restrict__ T,
                            const float* __restrict__ q,  const float* __restrict__ ps,
                            const float* __restrict__ a_int, const float* __restrict__ b_int,
                            float* __restrict__ Ain) {
  int t = blockIdx.x * blockDim.x + threadIdx.x;
  if (t >= HD * WD) return;
  int h = t / WD, w = t % WD;
  float p = ps[t];
  float pint[KD + 1], dlnp[KD], Tl[KD];
#pragma unroll
  for (int k = 0; k <= KD; ++k) pint[k] = a_int[k] + b_int[k] * p;
#pragma unroll
  for (int k = 0; k < KD; ++k) {
    Tl[k]   = T[((size_t)k * HD + h) * WD + w];
    dlnp[k] = __logf(pint[k + 1] / pint[k]);
  }
  float phi_int[KD + 1];
  phi_int[KD] = 0.f; // PHI_SURF
#pragma unroll
  for (int k = KD - 1; k >= 0; --k) phi_int[k] = phi_int[k + 1] + R_GAS_C * Tl[k] * dlnp[k];
#pragma unroll
  for (int k = 0; k < KD; ++k) {
    float u = uv[((size_t)(k * 2 + 0) * HD + h) * WD + w];
    float v = uv[((size_t)(k * 2 + 1) * HD + h) * WD + w];
    float ke = 0.5f * (u * u + v * v);
    Ain[((size_t)(0  + k) * HD + h) * WD + w] = 0.5f * (phi_int[k] + phi_int[k + 1]) + ke;
    Ain[((size_t)(8  + k) * HD + h) * WD + w] = Tl[k];
    Ain[((size_t)(16 + k) * HD + h) * WD + w] = q[((size_t)k * HD + h) * WD + w];
    Ain[((size_t)(25 + k) * HD + h) * WD + w] = u;
    Ain[((size_t)(33 + k) * HD + h) * WD + w] = v;
  }
  Ain[((size_t)24 * HD + h) * WD + w] = p;
}

// ----------------------------------------------------------------------------
// FP32 WMMA GEMM: C(MxN) = A(MxK) @ B(KxN). Each wave computes a 64x16
// supertile: one B fragment (single b64 load from the pair-interleaved B)
// feeds 4 independent v_wmma_f32_16x16x4_f32 chains (hides WMMA RAW latency).
// All loop addressing is strength-reduced to pointer increments (no per-iter
// 64-bit multiplies). A frag per ISA layout: lane<16 -> K {kk,kk+1},
// lane>=16 -> {kk+2,kk+3}. B frag: VGPR0 rows {kk,kk+2}, VGPR1 rows
// {kk+1,kk+3} by lane half. 4 waves / 128-thread block; A streamed with
// global_prefetch (B stays resident in L2: <0.6 MB << 192 MB).
// ----------------------------------------------------------------------------
__global__ void wmma_gemm_f32(const float* __restrict__ A, const float* __restrict__ Bp,
                              float* __restrict__ C, int K4, int lda, int ldb, int ldc) {
  int lane = threadIdx.x & 31;
  int wid  = threadIdx.x >> 5;
  int tn   = blockIdx.x;
  int tm4  = blockIdx.y * 4 + wid;       // 64-row supertile index
  int colB = tn * 16 + (lane & 15);
  int kh   = lane >> 4;                  // 0 / 1 -> K base offset 0 / 2
  const float* Ap0 = A + (size_t)(tm4 * 64 + (lane & 15)) * lda + (kh << 1);
  const float* Ap1 = Ap0 + (size_t)16 * lda;
  const float* Ap2 = Ap0 + (size_t)32 * lda;
  const float* Ap3 = Ap0 + (size_t)48 * lda;
  const v2f*   Bq  = (const v2f*)Bp + (size_t)kh * ldb + colB;
  const size_t bstep = (size_t)2 * ldb;  // pairs per k-step of 4
  v8f acc0 = {}, acc1 = {}, acc2 = {}, acc3 = {};
#pragma unroll 2
  for (int it = 0; it < K4; ++it) {
    v2f b  = *Bq;
    v2f a0 = *(const v2f*)Ap0;
    v2f a1 = *(const v2f*)Ap1;
    v2f a2 = *(const v2f*)Ap2;
    v2f a3 = *(const v2f*)Ap3;
    __builtin_prefetch(Ap0 + 128, 0, 3);
    __builtin_prefetch(Ap2 + 128, 0, 3);
    acc0 = __builtin_amdgcn_wmma_f32_16x16x4_f32(false, a0, false, b, (short)0, acc0, false, false);
    acc1 = __builtin_amdgcn_wmma_f32_16x16x4_f32(false, a1, false, b, (short)0, acc1, false, false);
    acc2 = __builtin_amdgcn_wmma_f32_16x16x4_f32(false, a2, false, b, (short)0, acc2, false, false);
    acc3 = __builtin_amdgcn_wmma_f32_16x16x4_f32(false, a3, false, b, (short)0, acc3, false, false);
    Bq  += bstep;
    Ap0 += 4; Ap1 += 4; Ap2 += 4; Ap3 += 4;
  }
  float* Cp = C + (size_t)(tm4 * 64 + (kh << 3)) * ldc + tn * 16 + (lane & 15);
#pragma unroll
  for (int j = 0; j < 8; ++j) Cp[(size_t)j * ldc] = acc0[j];
  Cp += (size_t)16 * ldc;
#pragma unroll
  for (int j = 0; j < 8; ++j) Cp[(size_t)j * ldc] = acc1[j];
  Cp += (size_t)16 * ldc;
#pragma unroll
  for (int j = 0; j < 8; ++j) Cp[(size_t)j * ldc] = acc2[j];
  Cp += (size_t)16 * ldc;
#pragma unroll
  for (int j = 0; j < 8; ++j) Cp[(size_t)j * ldc] = acc3[j];
}

// ----------------------------------------------------------------------------
// 7-tap gathered complex combine in spectral space (k already collapsed).
// Output rows: grad gc=c*2+d (0..49), vec 50+c*2+d (zeta d=0, div d=1), lap 66+c.
// m in [181,183] zero-fills the K-padding columns 362..367 for the irfft GEMM.
// All complex pairs accessed as aligned float2.
// ----------------------------------------------------------------------------
__global__ void spec_combine(const float2* __restrict__ G2, const int* __restrict__ gidx,
                             const float2* __restrict__ pV, const float2* __restrict__ pVV,
                             const float2* __restrict__ pS, float2* __restrict__ Y2) {
  int t = blockIdx.x * blockDim.x + threadIdx.x;
  if (t >= NCOUT * HD * LDF2) return;
  int m  = t % LDF2;
  int h  = (t / LDF2) % HD;
  int oc = t / (LDF2 * HD);
  size_t orow = ((size_t)oc * HD + h) * LDF2 + m;
  if (m > 180) { Y2[orow] = make_float2(0.f, 0.f); return; }
  float aR = 0.f, aI = 0.f;
  if (oc < 50) {                       // disco_grad on sstack channel c, comp d
    int c = oc >> 1, d = oc & 1;
#pragma unroll
    for (int n = 0; n < NTAP; ++n) {
      int r = gidx[h * NTAP + n];
      float2 g = G2[((size_t)c * HD + r) * LDF2 + m];
      float2 wv = pV[(((size_t)h * NTAP + n) * MD + m) * 2 + d];
      aR += g.x * wv.x - g.y * wv.y; aI += g.x * wv.y + g.y * wv.x;
    }
  } else if (oc < 66) {                // disco_vector on uv, output comp d
    int vv = oc - 50, c = vv >> 1, d = vv & 1;
#pragma unroll
    for (int n = 0; n < NTAP; ++n) {
      int r = gidx[h * NTAP + n];
      size_t wb0 = (((size_t)h * NTAP + n) * MD + m) * 4 + d;
#pragma unroll
      for (int e = 0; e < 2; ++e) {
        float2 g = G2[((size_t)(25 + 8 * e + c) * HD + r) * LDF2 + m];
        float2 wv = pVV[wb0 + e * 2];
        aR += g.x * wv.x - g.y * wv.y; aI += g.x * wv.y + g.y * wv.x;
      }
    }
  } else {                             // laplacian channels (T,q,u,v)
    int c = oc - 66;
    int src = (c < 16) ? (8 + c) : (9 + c);
#pragma unroll
    for (int n = 0; n < NTAP; ++n) {
      int r = gidx[h * NTAP + n];
      float2 g = G2[((size_t)src * HD + r) * LDF2 + m];
      float2 wv = pS[((size_t)h * NTAP + n) * MD + m];
      aR += g.x * wv.x - g.y * wv.y; aI += g.x * wv.y + g.y * wv.x;
    }
  }
  Y2[orow] = make_float2(aR, aI);
}

// ----------------------------------------------------------------------------
// Vertical physics per (h,w) column: cumsums over K=8, vertical advection,
// Coriolis/PGF, tendencies. Out channels: grad 0..49, zeta/div 50..65, lap 66..97.
// ----------------------------------------------------------------------------
__global__ void final_step(const float* __restrict__ uv, const float* __restrict__ T,
                           const float* __restrict__ q,  const float* __restrict__ ps,
                           const float* __restrict__ a_int, const float* __restrict__ b_int,
                           const float* __restrict__ a_mid, const float* __restrict__ b_mid,
                           const float* __restrict__ f_cor, const float* __restrict__ Out,
                           float* __restrict__ dOut) {
  int t = blockIdx.x * blockDim.x + threadIdx.x;
  if (t >= HD * WD) return;
  int h = t / WD, w = t % WD;
  auto OC = [&](int oc) -> float { return Out[((size_t)oc * HD + h) * LDF + w]; };
  float p = ps[t];
  float pint[KD + 1], dp[KD], pmid[KD], Tl[KD], ul[KD], vl[KD], ql[KD];
#pragma unroll
  for (int k = 0; k <= KD; ++k) pint[k] = a_int[k] + b_int[k] * p;
#pragma unroll
  for (int k = 0; k < KD; ++k) {
    dp[k]   = pint[k + 1] - pint[k];
    pmid[k] = a_mid[k] + b_mid[k] * p;
    Tl[k] = T[((size_t)k * HD + h) * WD + w];
    ql[k] = q[((size_t)k * HD + h) * WD + w];
    ul[k] = uv[((size_t)(k * 2 + 0) * HD + h) * WD + w];
    vl[k] = uv[((size_t)(k * 2 + 1) * HD + h) * WD + w];
  }
  float gps0 = OC(48), gps1 = OC(49);
  float vdg[KD], dflux[KD], Ccum[KD + 1];
  Ccum[0] = 0.f;
#pragma unroll
  for (int k = 0; k < KD; ++k) {
    vdg[k]   = ul[k] * gps0 + vl[k] * gps1;
    dflux[k] = OC(51 + 2 * k) * dp[k] + (b_int[k + 1] - b_int[k]) * vdg[k];
    Ccum[k + 1] = Ccum[k] + dflux[k];
  }
  float total = Ccum[KD];
  float edot[KD + 1], omg[KD];
#pragma unroll
  for (int k = 0; k <= KD; ++k) edot[k] = b_int[k] * total - Ccum[k];
#pragma unroll
  for (int k = 0; k < KD; ++k) omg[k] = b_mid[k] * vdg[k] - (Ccum[k] + 0.5f * dflux[k]);
  auto vadv = [&](const float* X, float* o) {
    float dd[KD + 1];
    dd[0] = 0.f; dd[KD] = 0.f;
#pragma unroll
    for (int j = 1; j < KD; ++j) dd[j] = X[j] - X[j - 1];
#pragma unroll
    for (int k = 0; k < KD; ++k)
      o[k] = -(edot[k] * dd[k] + edot[k + 1] * dd[k + 1]) / (2.f * dp[k]);
  };
  float vT[KD], vq[KD], vu[KD], vv[KD];
  vadv(Tl, vT); vadv(ql, vq); vadv(ul, vu); vadv(vl, vv);
  float fc = f_cor[h];
#pragma unroll
  for (int k = 0; k < KD; ++k) {
    float av  = OC(50 + 2 * k) + fc;                 // zeta + f
    float pgf = -(R_GAS_C * Tl[k] * b_mid[k] / pmid[k]);
    float du  =  av * vl[k] - OC(2 * k)     + pgf * gps0 + vu[k] + OC(82 + k);
    float dv  = -av * ul[k] - OC(2 * k + 1) + pgf * gps1 + vv[k] + OC(90 + k);
    float advT = -(ul[k] * OC(16 + 2 * k) + vl[k] * OC(17 + 2 * k));
    float advq = -(ul[k] * OC(32 + 2 * k) + vl[k] * OC(33 + 2 * k));
    float dT = advT + (R_GAS_C / C_P_C) * Tl[k] * omg[k] / pmid[k] + vT[k] + OC(66 + k);
    float dq = advq + vq[k] + OC(74 + k);
    dOut[((size_t)(2 * k + 0) * HD + h) * WD + w] = du;
    dOut[((size_t)(2 * k + 1) * HD + h) * WD + w] = dv;
    dOut[((size_t)(16 + k)    * HD + h) * WD + w] = dT;
    dOut[((size_t)(24 + k)    * HD + h) * WD + w] = dq;
  }
  dOut[((size_t)32 * HD + h) * WD + w] = -total;
}

// ----------------------------------------------------------------------------
extern "C" void kernel_launch(void* const* d_in, const int* in_sizes, int n_in,
                              void* d_out, int out_size, void* d_ws, size_t ws_size,
                              hipStream_t stream) {
  const float*  uv     = (const float*) d_in[0];
  const float*  T      = (const float*) d_in[1];
  const float*  q      = (const float*) d_in[2];
  const float*  ps     = (const float*) d_in[3];
  const float2* psi_s  = (const float2*)d_in[4];
  const float2* psi_v  = (const float2*)d_in[5];
  const float2* psi_vv = (const float2*)d_in[6];
  const int*    gidx   = (const int*)   d_in[7];
  const float*  lap_w  = (const float*) d_in[8];
  const float*  a_mid  = (const float*) d_in[9];
  const float*  b_mid  = (const float*) d_in[10];
  const float*  a_int  = (const float*) d_in[11];
  const float*  b_int  = (const float*) d_in[12];
  const float*  f_cor  = (const float*) d_in[13];
  if (ws_size < WS_FLOATS * sizeof(float)) return; // workspace too small

  float* ws   = (float*)d_ws;
  float* FWD  = ws + OFF_FWD;
  float* INV  = ws + OFF_INV;
  float* AIN  = ws + OFF_AIN;
  float* GHAT = ws + OFF_GHAT;
  float* YS   = ws + OFF_YS;
  float* OUTS = ws + OFF_OUTS;
  float2* PV  = (float2*)(ws + OFF_PV);
  float2* PVV = (float2*)(ws + OFF_PVV);
  float2* PS  = (float2*)(ws + OFF_PS);

  const int TB = 256;
  gen_dft<<<(368 * 368 + TB - 1) / TB, TB, 0, stream>>>(FWD, INV);
  psi_combine<<<(HD * NTAP * MD + TB - 1) / TB, TB, 0, stream>>>(
      psi_s, psi_v, psi_vv, lap_w, PV, PVV, PS);
  pack_fields<<<(HD * WD + TB - 1) / TB, TB, 0, stream>>>(
      uv, T, q, ps, a_int, b_int, AIN);
  // forward rfft as GEMM: (7424 x 360) @ (360 x 368); 64x16 per wave
  wmma_gemm_f32<<<dim3(LDF / 16, MROWS_F / 256), 128, 0, stream>>>(
      AIN, FWD, GHAT, 360 / 4, 360, LDF, LDF);
  spec_combine<<<(NCOUT * HD * LDF2 + TB - 1) / TB, TB, 0, stream>>>(
      (const float2*)GHAT, gidx, PV, PVV, PS, (float2*)YS);
  // irfft as GEMM: (17664 x 368) @ (368 x 368); 64x16 per wave
  wmma_gemm_f32<<<dim3(LDF / 16, MROWS_I / 256), 128, 0, stream>>>(
      YS, INV, OUTS, LDF / 4, LDF, LDF, LDF);
  final_step<<<(HD * WD + TB - 1) / TB, TB, 0, stream>>>(
      uv, T, q, ps, a_int, b_int, a_mid, b_mid, f_cor, OUTS, (float*)d_out);
}